// FlaxBloomAttention_80058190397627
// MI455X (gfx1250) — compile-verified
//
#include <hip/hip_runtime.h>

typedef __attribute__((ext_vector_type(2))) float v2f;
typedef __attribute__((ext_vector_type(8))) float v8f;

constexpr int Bc = 2, Sc = 2048, Hc = 4096, NHc = 32, HDc = 128;
constexpr float MASK_VAL = -3.0e38f;

__device__ __forceinline__ v8f wmma_f32(v2f a, v2f b, v8f c) {
  // D = A(16x4) * B(4x16) + C(16x16), fp32 tensor path on gfx1250
  return __builtin_amdgcn_wmma_f32_16x16x4_f32(false, a, false, b, (short)0, c,
                                               false, false);
}

// gfx1250 async global->LDS copy (ASYNCcnt class, no VGPR staging).
// VDST carries the LDS byte offset (low 32 bits of the flat LDS address).
__device__ __forceinline__ void async_copy16(void* lds_dst, const void* gsrc) {
  asm volatile("global_load_async_to_lds_b128 %0, %1, off" ::"v"(
                   (unsigned)(uintptr_t)lds_dst),
               "v"(gsrc)
               : "memory");
}
__device__ __forceinline__ void wait_async0() {
  asm volatile("s_wait_asynccnt 0x0" ::: "memory");
}
__device__ __forceinline__ void wait_ds0() {
  asm volatile("s_wait_dscnt 0x0" ::: "memory");
}

// ---------------------------------------------------------------------------
// Tiled GEMM: C[M,N] = A[M,K] @ Bm[K,N] + bias[N]
// Block: 256 thr = 8 waves (2x4 wave grid), block tile 64x128, K-tile 32.
// Each wave owns a 32x32 register tile (2x2 WMMA accumulators) so every
// A/B fragment fetched from LDS feeds two v_wmma_f32_16x16x4_f32.
// Global->LDS staging uses global_load_async_to_lds_b128.
// MODE 0: QKV epilogue — decode n -> (head, {q,k,v}, d), scatter into
//         Q/K/V workspace laid out [B, NH, S, HD]; Q pre-scaled by 1/sqrt(HD).
// MODE 1: plain bias-add store (output projection).
// ---------------------------------------------------------------------------
template <int MODE>
__global__ __launch_bounds__(256) void gemm_wmma_kernel(
    const float* __restrict__ A, const float* __restrict__ Bm,
    const float* __restrict__ bias, float* __restrict__ C,
    float* __restrict__ Qo, float* __restrict__ Ko, float* __restrict__ Vo,
    int M, int N, int K, float qscale) {
  constexpr int TM = 64, TN = 128, TK = 32;
  __shared__ __align__(16) float As[TM * TK];  // 8 KB
  __shared__ __align__(16) float Bs[TK * TN];  // 16 KB

  const int tid = threadIdx.x;
  const int wave = tid >> 5, lane = tid & 31;
  const int lr = lane & 15, lh = lane >> 4;
  const int wm = wave >> 2, wn = wave & 3;  // wave tile origin (wm*32, wn*32)
  const int m0 = blockIdx.y * TM;
  const int n0 = blockIdx.x * TN;

  v8f acc[2][2] = {};

  for (int kb = 0; kb < K; kb += TK) {
    // In-flight DS reads must drain before async engine rewrites LDS
    // (DScnt and ASYNCcnt are independent counter classes).
    wait_ds0();
#pragma unroll
    for (int i = 0; i < 2; ++i) {  // A tile 64x32: 2 x b128 per thread
      int e = (tid + i * 256) * 4;
      int r = e >> 5, c = e & 31;
      async_copy16(&As[r * TK + c], &A[(size_t)(m0 + r) * K + kb + c]);
    }
#pragma unroll
    for (int i = 0; i < 4; ++i) {  // B tile 32x128: 4 x b128 per thread
      int e = (tid + i * 256) * 4;
      int r = e >> 7, c = e & 127;
      async_copy16(&Bs[r * TN + c], &Bm[(size_t)(kb + r) * N + n0 + c]);
    }
    wait_async0();
    __syncthreads();
#pragma unroll
    for (int kk = 0; kk < TK; kk += 4) {
      v2f a0 = *(const v2f*)&As[(wm * 32 + lr) * TK + kk + lh * 2];
      v2f a1 = *(const v2f*)&As[(wm * 32 + 16 + lr) * TK + kk + lh * 2];
      v2f b0, b1;
      b0.x = Bs[(kk + lh * 2 + 0) * TN + wn * 32 + lr];
      b0.y = Bs[(kk + lh * 2 + 1) * TN + wn * 32 + lr];
      b1.x = Bs[(kk + lh * 2 + 0) * TN + wn * 32 + 16 + lr];
      b1.y = Bs[(kk + lh * 2 + 1) * TN + wn * 32 + 16 + lr];
      acc[0][0] = wmma_f32(a0, b0, acc[0][0]);
      acc[0][1] = wmma_f32(a0, b1, acc[0][1]);
      acc[1][0] = wmma_f32(a1, b0, acc[1][0]);
      acc[1][1] = wmma_f32(a1, b1, acc[1][1]);
    }
    __syncthreads();
  }

#pragma unroll
  for (int tn = 0; tn < 2; ++tn) {
    const int col = n0 + wn * 32 + tn * 16 + lr;
    const float bv = bias[col];
    if (MODE == 1) {
#pragma unroll
      for (int tm = 0; tm < 2; ++tm) {
#pragma unroll
        for (int r = 0; r < 8; ++r) {
          int row = m0 + wm * 32 + tm * 16 + lh * 8 + r;
          C[(size_t)row * N + col] = acc[tm][tn][r] + bv;
        }
      }
    } else {
      // n indexes (head, third, d): w_qkv is [H, NH, 3*HD]
      const int h = col / (3 * HDc);
      const int rem = col % (3 * HDc);
      const int t = rem / HDc;
      const int d = rem % HDc;
      float* dst = (t == 0) ? Qo : (t == 1) ? Ko : Vo;
      const float sc = (t == 0) ? qscale : 1.0f;
#pragma unroll
      for (int tm = 0; tm < 2; ++tm) {
#pragma unroll
        for (int r = 0; r < 8; ++r) {
          int row = m0 + wm * 32 + tm * 16 + lh * 8 + r;  // token b*S + s
          int bb = row / Sc, ss = row % Sc;
          dst[(((size_t)bb * NHc + h) * Sc + ss) * HDc + d] =
              (acc[tm][tn][r] + bv) * sc;
        }
      }
    }
  }
}

// ---------------------------------------------------------------------------
// Flash-style attention. One wave owns one 16-row query tile of one (b,h).
// Q,K,V are [B,NH,S,HD]; O (attn output) is [B,S,NH,HD] so the output
// projection sees a row-major [B*S, NH*HD] matrix.
// ---------------------------------------------------------------------------
__global__ __launch_bounds__(256) void attn_kernel(
    const float* __restrict__ Q, const float* __restrict__ K,
    const float* __restrict__ V, const float* __restrict__ alibi,
    const int* __restrict__ amask, float* __restrict__ O) {
  __shared__ __align__(16) float Pshare[8][256];  // per-wave 16x16 P scratch
  const int tid = threadIdx.x;
  const int wave = tid >> 5, lane = tid & 31;
  const int lr = lane & 15, lh = lane >> 4;

  const int gw = blockIdx.x * 8 + wave;  // global wave id
  const int qt = gw & 127;               // S/16 = 128 query tiles
  const int bh = gw >> 7;                // 0..B*NH-1
  const int b = bh >> 5, h = bh & 31;
  const int q0 = qt * 16;

  const float* Qb = Q + (size_t)bh * Sc * HDc;
  const float* Kb = K + (size_t)bh * Sc * HDc;
  const float* Vb = V + (size_t)bh * Sc * HDc;

  // Q tile as 32 A-fragments over HD=128 (already scaled by 1/sqrt(HD))
  v2f qf[32];
#pragma unroll
  for (int c = 0; c < 32; ++c)
    qf[c] = *(const v2f*)&Qb[(size_t)(q0 + lr) * HDc + c * 4 + lh * 2];

  v8f oacc[8] = {};
  float rmax[8], rsum[8];
#pragma unroll
  for (int r = 0; r < 8; ++r) {
    rmax[r] = MASK_VAL;
    rsum[r] = 0.f;
  }
  float* ps = &Pshare[wave][0];

  for (int kt = 0; kt <= qt; ++kt) {  // causal: only tiles with k0 <= q0+15
    const int k0 = kt * 16;
    v8f s = {};
#pragma unroll
    for (int c = 0; c < 32; ++c) {
      // B-frag: K^T[hd][key]: lane holds key=k0+lr, hd = c*4+lh*2+{0,1}
      v2f kf = *(const v2f*)&Kb[(size_t)(k0 + lr) * HDc + c * 4 + lh * 2];
      s = wmma_f32(qf[c], kf, s);
    }
    const float al = alibi[(size_t)bh * Sc + k0 + lr];  // [B,NH,1,S]
    const bool pad_ok = amask[b * Sc + k0 + lr] > 0;
#pragma unroll
    for (int r = 0; r < 8; ++r) {
      const int qrow = q0 + lh * 8 + r;
      const bool ok = pad_ok && (k0 + lr <= qrow);
      float sv = (ok ? s[r] : MASK_VAL) + al;
      // row max across the 16 lanes sharing this row group
      float m = sv;
      m = fmaxf(m, __shfl_xor(m, 1, 32));
      m = fmaxf(m, __shfl_xor(m, 2, 32));
      m = fmaxf(m, __shfl_xor(m, 4, 32));
      m = fmaxf(m, __shfl_xor(m, 8, 32));
      const float nm = fmaxf(rmax[r], m);
      const float corr = __expf(rmax[r] - nm);
      rmax[r] = nm;
      const float p = __expf(sv - nm);
      float psum = p;
      psum += __shfl_xor(psum, 1, 32);
      psum += __shfl_xor(psum, 2, 32);
      psum += __shfl_xor(psum, 4, 32);
      psum += __shfl_xor(psum, 8, 32);
      rsum[r] = rsum[r] * corr + psum;
#pragma unroll
      for (int t2 = 0; t2 < 8; ++t2) oacc[t2][r] *= corr;
      ps[(lh * 8 + r) * 16 + lr] = p;  // C-layout -> row-major LDS
    }
    // P(16x16) @ V(16x128); A-frags re-loaded transposed from LDS.
    // Same-wave DS ops are processed in order, so no barrier needed.
#pragma unroll
    for (int c4 = 0; c4 < 4; ++c4) {
      v2f pa = *(const v2f*)&ps[lr * 16 + c4 * 4 + lh * 2];
#pragma unroll
      for (int nt = 0; nt < 8; ++nt) {
        const float* vp =
            &Vb[(size_t)(k0 + c4 * 4 + lh * 2) * HDc + nt * 16 + lr];
        v2f vb;
        vb.x = vp[0];
        vb.y = vp[HDc];
        oacc[nt] = wmma_f32(pa, vb, oacc[nt]);
      }
    }
  }

#pragma unroll
  for (int nt = 0; nt < 8; ++nt) {
#pragma unroll
    for (int r = 0; r < 8; ++r) {
      const int qrow = q0 + lh * 8 + r;
      O[(((size_t)b * Sc + qrow) * NHc + h) * HDc + nt * 16 + lr] =
          oacc[nt][r] / rsum[r];
    }
  }
}

extern "C" void kernel_launch(void* const* d_in, const int* in_sizes, int n_in,
                              void* d_out, int out_size, void* d_ws,
                              size_t ws_size, hipStream_t stream) {
  const float* hs = (const float*)d_in[0];     // [B,S,H]
  const float* alibi = (const float*)d_in[1];  // [B,NH,1,S]
  const float* w_qkv = (const float*)d_in[2];  // [H,NH,3*HD]
  const float* b_qkv = (const float*)d_in[3];  // [NH,3*HD]
  const float* w_out = (const float*)d_in[4];  // [NH,HD,H] == [NH*HD, H]
  const float* b_out = (const float*)d_in[5];  // [H]
  const int* amask = (const int*)d_in[6];      // [B,S]
  float* out = (float*)d_out;                  // [B,S,H]

  const size_t per = (size_t)Bc * NHc * Sc * HDc;  // 16.7M floats
  float* Qp = (float*)d_ws;
  float* Kp = Qp + per;
  float* Vp = Kp + per;
  float* AO = Vp + per;  // attn output [B,S,NH,HD]

  const float qscale = 0.088388347648318447f;  // 1/sqrt(128)
  dim3 blk(256);

  // 1) fused QKV projection + scatter: M=B*S=4096, N=NH*3*HD=12288, K=H=4096
  gemm_wmma_kernel<0><<<dim3((NHc * 3 * HDc) / 128, (Bc * Sc) / 64), blk, 0,
                        stream>>>(hs, w_qkv, b_qkv, nullptr, Qp, Kp, Vp,
                                  Bc * Sc, NHc * 3 * HDc, Hc, qscale);

  // 2) causal alibi attention: B*NH*(S/16) waves / 8 waves per block
  attn_kernel<<<dim3((Bc * NHc * (Sc / 16)) / 8), blk, 0, stream>>>(
      Qp, Kp, Vp, alibi, amask, AO);

  // 3) output projection: M=B*S=4096, N=H=4096, K=NH*HD=4096
  gemm_wmma_kernel<1><<<dim3(Hc / 128, (Bc * Sc) / 64), blk, 0, stream>>>(
      AO, w_out, b_out, out, nullptr, nullptr, nullptr, Bc * Sc, Hc,
      NHc * HDc, 1.0f);
}